// AtLocPlusCriterion_VO_30597347017295
// MI455X (gfx1250) — compile-verified
//
#include <hip/hip_runtime.h>
#include <hip/hip_bf16.h>
#include <stdint.h>

#define TPB 256
#define PAIRS 256
#define NBLOCKS 2048
#define PRED_TILE_F (PAIRS * 6 + 8)   // 1544 floats of LDS (fetch 1542)
#define TARG_TILE_F (PAIRS * 6)       // 1536 floats
#define PRED_CHUNKS16 385             // 385 x b128 covers floats [0,1540)
#define TARG_CHUNKS16 384             // 384 x b128 covers floats [0,1536)
#define TOT_CHUNKS (PRED_CHUNKS16 + 1 + TARG_CHUNKS16) // +1 = b64 tail of pred

typedef int v4i __attribute__((vector_size(16)));
typedef int v2i __attribute__((vector_size(8)));
typedef __attribute__((address_space(1))) v4i* g4p;   // global v4i*
typedef __attribute__((address_space(3))) v4i* l4p;   // LDS v4i*
typedef __attribute__((address_space(1))) v2i* g2p;   // global v2i*
typedef __attribute__((address_space(3))) v2i* l2p;   // LDS v2i*

// ---- CDNA5 async global->LDS copies (ASYNCcnt-tracked DMA path) ----------
__device__ __forceinline__ void async_copy_b128(const float* gsrc, float* lds) {
#if __has_builtin(__builtin_amdgcn_global_load_async_to_lds_b128)
  __builtin_amdgcn_global_load_async_to_lds_b128(
      (g4p)(unsigned long long)(uintptr_t)gsrc,
      (l4p)(unsigned)(uintptr_t)lds, 0, 0);
#else
  asm volatile("global_load_async_to_lds_b128 %0, %1, off"
               :: "v"((unsigned)(uintptr_t)lds), "v"(gsrc) : "memory");
#endif
}

__device__ __forceinline__ void async_copy_b64(const float* gsrc, float* lds) {
#if __has_builtin(__builtin_amdgcn_global_load_async_to_lds_b64)
  __builtin_amdgcn_global_load_async_to_lds_b64(
      (g2p)(unsigned long long)(uintptr_t)gsrc,
      (l2p)(unsigned)(uintptr_t)lds, 0, 0);
#else
  asm volatile("global_load_async_to_lds_b64 %0, %1, off"
               :: "v"((unsigned)(uintptr_t)lds), "v"(gsrc) : "memory");
#endif
}

__device__ __forceinline__ void wait_async0() {
#if __has_builtin(__builtin_amdgcn_s_wait_asynccnt)
  __builtin_amdgcn_s_wait_asynccnt(0);
#else
  asm volatile("s_wait_asynccnt 0" ::: "memory");
#endif
}

// ---- per-pair VO math (all fp32, matches reference semantics) ------------
__device__ __forceinline__ void qexp4(const float v[3], float q[4]) {
  float n2 = v[0] * v[0] + v[1] * v[1] + v[2] * v[2];
  float n  = fmaxf(sqrtf(n2), 1e-8f);
  float s, c;
  sincosf(n, &s, &c);
  float k = s / n;
  q[0] = c; q[1] = v[0] * k; q[2] = v[1] * k; q[3] = v[2] * k;
}

// r = t + 2*qs*cross(qv,t) + 2*cross(qv, cross(qv,t))
__device__ __forceinline__ void rotv(const float t[3], float qs, const float qv[3], float r[3]) {
  float b0 = qv[1] * t[2] - qv[2] * t[1];
  float b1 = qv[2] * t[0] - qv[0] * t[2];
  float b2 = qv[0] * t[1] - qv[1] * t[0];
  float c0 = qv[1] * b2 - qv[2] * b1;
  float c1 = qv[2] * b0 - qv[0] * b2;
  float c2 = qv[0] * b1 - qv[1] * b0;
  r[0] = t[0] + 2.0f * (b0 * qs + c0);
  r[1] = t[1] + 2.0f * (b1 * qs + c1);
  r[2] = t[2] + 2.0f * (b2 * qs + c2);
}

__device__ __forceinline__ void vo_pair(const float* __restrict__ p0,
                                        const float* __restrict__ p1,
                                        const float* __restrict__ tg,
                                        float& ts_acc, float& qs_acc) {
  float t0[3] = {p0[0], p0[1], p0[2]};
  float w0[3] = {p0[3], p0[4], p0[5]};
  float t1[3] = {p1[0], p1[1], p1[2]};
  float w1[3] = {p1[3], p1[4], p1[5]};
  float q0[4], q1[4];
  qexp4(w0, q0);
  qexp4(w1, q1);
  float s0 = q0[0];
  float v0[3] = {-q0[1], -q0[2], -q0[3]};          // q0_inv vector part
  float r0[3];
  rotv(t0, s0, v0, r0);
  float ti[3] = {-r0[0], -r0[1], -r0[2]};          // t0_inv
  float s1 = q1[0];
  float v1[3] = {q1[1], q1[2], q1[3]};
  // q = normalize(qmult(q0_inv, q1))
  float qs = s0 * s1 - (v0[0] * v1[0] + v0[1] * v1[1] + v0[2] * v1[2]);
  float qv[3];
  qv[0] = v0[0] * s1 + v1[0] * s0 + (v0[1] * v1[2] - v0[2] * v1[1]);
  qv[1] = v0[1] * s1 + v1[1] * s0 + (v0[2] * v1[0] - v0[0] * v1[2]);
  qv[2] = v0[2] * s1 + v1[2] * s0 + (v0[0] * v1[1] - v0[1] * v1[0]);
  float qn  = sqrtf(qs * qs + qv[0] * qv[0] + qv[1] * qv[1] + qv[2] * qv[2]);
  float inv = 1.0f / qn;
  qs *= inv; qv[0] *= inv; qv[1] *= inv; qv[2] *= inv;
  // t = t0_inv + rot(t1, q0_inv)
  float r1[3];
  rotv(t1, s0, v0, r1);
  float tx = ti[0] + r1[0], ty = ti[1] + r1[1], tz = ti[2] + r1[2];
  // qlog
  float nv  = fmaxf(sqrtf(qv[0] * qv[0] + qv[1] * qv[1] + qv[2] * qv[2]), 1e-8f);
  float ang = acosf(fminf(fmaxf(qs, -1.0f), 1.0f));
  float kk  = ang / nv;
  float l0 = qv[0] * kk, l1 = qv[1] * kk, l2 = qv[2] * kk;
  ts_acc += fabsf(tx - tg[0]) + fabsf(ty - tg[1]) + fabsf(tz - tg[2]);
  qs_acc += fabsf(l0 - tg[3]) + fabsf(l1 - tg[4]) + fabsf(l2 - tg[5]);
}

// ---- kernel 1: per-block partial sums via LDS-staged tiles ---------------
__global__ __launch_bounds__(TPB) void vo_loss_partial(
    const float* __restrict__ pred, const float* __restrict__ targ,
    long nPairs, int numTiles,
    float* __restrict__ wsT, float* __restrict__ wsQ) {
  __shared__ __align__(16) float sPred[PRED_TILE_F];
  __shared__ __align__(16) float sTarg[TARG_TILE_F];
  __shared__ float redT[TPB];
  __shared__ float redQ[TPB];
  const int tid = threadIdx.x;
  float ts = 0.0f, qs = 0.0f;

  for (int tile = blockIdx.x; tile < numTiles; tile += gridDim.x) {
    const long base = (long)tile * PAIRS;
    const bool full = (base + PAIRS) <= nPairs;   // uniform across block
    if (full) {
      __syncthreads();                            // LDS reuse safety
      const float* gp = pred + base * 6;          // 16B-aligned (base even*24B)
      const float* gt = targ + base * 6;
      for (int c = tid; c < TOT_CHUNKS; c += TPB) {
        if (c < PRED_CHUNKS16) {
          async_copy_b128(gp + c * 4, &sPred[c * 4]);
        } else if (c == PRED_CHUNKS16) {
          async_copy_b64(gp + PRED_CHUNKS16 * 4, &sPred[PRED_CHUNKS16 * 4]);
        } else {
          const int cc = c - (PRED_CHUNKS16 + 1);
          async_copy_b128(gt + cc * 4, &sTarg[cc * 4]);
        }
      }
      wait_async0();                              // my wave's DMAs landed in LDS
      __syncthreads();                            // everyone's DMAs visible
      vo_pair(&sPred[tid * 6], &sPred[tid * 6 + 6], &sTarg[tid * 6], ts, qs);
    } else {
      const long p = base + tid;
      if (p < nPairs)
        vo_pair(pred + p * 6, pred + p * 6 + 6, targ + p * 6, ts, qs);
    }
  }

  __syncthreads();
  redT[tid] = ts;
  redQ[tid] = qs;
  __syncthreads();
  for (int s = TPB / 2; s > 0; s >>= 1) {
    if (tid < s) { redT[tid] += redT[tid + s]; redQ[tid] += redQ[tid + s]; }
    __syncthreads();
  }
  if (tid == 0) { wsT[blockIdx.x] = redT[0]; wsQ[blockIdx.x] = redQ[0]; }
}

// ---- kernel 2: deterministic final combine -------------------------------
__global__ __launch_bounds__(TPB) void vo_loss_final(
    const float* __restrict__ wsT, const float* __restrict__ wsQ,
    const float* __restrict__ srx, const float* __restrict__ srq,
    float* __restrict__ out, int nblocks, long nPairs) {
  __shared__ double rT[TPB];
  __shared__ double rQ[TPB];
  const int tid = threadIdx.x;
  double a = 0.0, b = 0.0;
  for (int i = tid; i < nblocks; i += TPB) { a += (double)wsT[i]; b += (double)wsQ[i]; }
  rT[tid] = a; rQ[tid] = b;
  __syncthreads();
  for (int s = TPB / 2; s > 0; s >>= 1) {
    if (tid < s) { rT[tid] += rT[tid + s]; rQ[tid] += rQ[tid + s]; }
    __syncthreads();
  }
  if (tid == 0) {
    const double denom = 3.0 * (double)nPairs;
    const float tl = (float)(rT[0] / denom);
    const float ql = (float)(rQ[0] / denom);
    const float sx = srx[0];
    const float sq = srq[0];
    out[0] = expf(-sx) * tl + sx + expf(-sq) * ql + sq;
  }
}

extern "C" void kernel_launch(void* const* d_in, const int* in_sizes, int n_in,
                              void* d_out, int out_size, void* d_ws, size_t ws_size,
                              hipStream_t stream) {
  const float* pred = (const float*)d_in[0];
  const float* targ = (const float*)d_in[1];
  const float* srx  = (const float*)d_in[2];
  const float* srq  = (const float*)d_in[3];
  const long nPairs = (long)in_sizes[1] / 6;       // targ rows = T-1
  int numTiles = (int)((nPairs + PAIRS - 1) / PAIRS);
  if (numTiles < 1) numTiles = 1;
  int grid = NBLOCKS < numTiles ? NBLOCKS : numTiles;

  float* wsT = (float*)d_ws;
  float* wsQ = wsT + grid;

  vo_loss_partial<<<grid, TPB, 0, stream>>>(pred, targ, nPairs, numTiles, wsT, wsQ);
  vo_loss_final<<<1, TPB, 0, stream>>>(wsT, wsQ, srx, srq, (float*)d_out, grid, nPairs);
}